// GraphCF_68599217652446
// MI455X (gfx1250) — compile-verified
//
#include <hip/hip_runtime.h>
#include <cstdint>

typedef __attribute__((ext_vector_type(2))) float v2f;
typedef __attribute__((ext_vector_type(8))) float v8f;

#define NND 50000
#define NE  600000
#define DD  128
#define BB  128

__device__ inline v8f wmma_f32_k4(v2f a, v2f b, v8f c) {
  return __builtin_amdgcn_wmma_f32_16x16x4_f32(false, a, false, b, (short)0, c, false, false);
}

// ---------------- elementwise / setup kernels ----------------
__global__ void k_fill(float* p, long n, float v) {
  long i = (long)blockIdx.x * blockDim.x + threadIdx.x;
  if (i < n) p[i] = v;
}

__global__ void k_edge_deg(const int* __restrict__ dst, float* __restrict__ deg, int ne) {
  int e = blockIdx.x * blockDim.x + threadIdx.x;
  if (e < ne) atomicAdd(&deg[dst[e]], 1.0f);
}

__global__ void k_rsqrt(float* p, int n) {
  int i = blockIdx.x * blockDim.x + threadIdx.x;
  if (i < n) p[i] = rsqrtf(p[i]);
}

// ---------------- strip-mined WMMA GEMM ----------------
// Out strip per wave: 16 rows x (NT*16) cols. Block = 8 waves = 128 rows.
// Weight panel staged in LDS in <=128-row chunks (64 KB).
// Epilogue writes OutRaw (=hw) and/or Out2 = raw*dinv[row]^2 + bias (relu opt).
template<int NT>
__global__ void k_wmma_gemm_strip(const float* __restrict__ A, int lda,
                                  const int* __restrict__ rowIdx,
                                  const float* __restrict__ Bm,   // [K, NT*16] row-major
                                  float* __restrict__ OutRaw,     // nullable, ld = NT*16
                                  float* __restrict__ Out2, int ldo2,  // nullable
                                  const float* __restrict__ dinv, // nullable -> scale 1
                                  const float* __restrict__ bias, // used if Out2
                                  int relu, int K, int M) {
  extern __shared__ float smem[];
  const int F = NT * 16;
  int tid = threadIdx.x;
  int wave = tid >> 5, lane = tid & 31;
  int half = lane >> 4, lm = lane & 15;
  long mt = (long)blockIdx.x * 8 + wave;
  bool active = (mt * 16) < (long)M;
  int m0 = (int)(mt << 4);
  const float2* ap = nullptr;
  if (active) {
    int r = m0 + lm;
    long ar = rowIdx ? (long)rowIdx[r] : (long)r;
    ap = (const float2*)(A + ar * (long)lda);
  }
  v8f zero = {0.f,0.f,0.f,0.f,0.f,0.f,0.f,0.f};
  v8f acc[NT];
  #pragma unroll
  for (int nt = 0; nt < NT; ++nt) acc[nt] = zero;

  for (int kc = 0; kc < K; kc += 128) {
    int KC = (K - kc) < 128 ? (K - kc) : 128;
    const float4* g4 = (const float4*)(Bm + (long)kc * F);
    float4* l4 = (float4*)smem;
    int tot4 = (KC * F) >> 2;
    for (int i = tid; i < tot4; i += 256) l4[i] = g4[i];
    __syncthreads();
    if (active) {
      for (int k = 0; k < KC; k += 4) {
        int ka = k + (half << 1);
        float2 av = ap[(kc + ka) >> 1];
        v2f a; a.x = av.x; a.y = av.y;
        #pragma unroll
        for (int nt = 0; nt < NT; ++nt) {
          int n = (nt << 4) + lm;
          v2f b; b.x = smem[ka * F + n]; b.y = smem[(ka + 1) * F + n];
          acc[nt] = wmma_f32_k4(a, b, acc[nt]);
        }
      }
    }
    __syncthreads();
  }
  if (!active) return;
  #pragma unroll
  for (int v = 0; v < 8; ++v) {
    int row = m0 + v + (half << 3);
    float sc = dinv ? dinv[row] * dinv[row] : 1.0f;
    #pragma unroll
    for (int nt = 0; nt < NT; ++nt) {
      int n = (nt << 4) + lm;
      float raw = acc[nt][v];
      if (OutRaw) OutRaw[(long)row * F + n] = raw;
      if (Out2) {
        float d2 = raw * sc + bias[n];
        if (relu) d2 = fmaxf(d2, 0.0f);
        Out2[(long)row * ldo2 + n] = d2;
      }
    }
  }
}

// ---------------- strip-mined implicit-GEMM conv1d ----------------
// channel-last: Y[b,l,co] = relu(sum_t,ci X[b,l+t,ci]*W[co,ci,t] + bias)
// per tap: stage W[:,:,t] transposed into LDS as [ci*Cout + co]
template<int NT>
__global__ void k_wmma_conv_strip(const float* __restrict__ X,
                                  const float* __restrict__ Wt,  // [Cout,Cin,T]
                                  const float* __restrict__ bias,
                                  float* __restrict__ Y,
                                  int Lin, int Lout, int Cin, int T, int M) {
  extern __shared__ float smem[];
  const int Cout = NT * 16;
  int tid = threadIdx.x;
  int wave = tid >> 5, lane = tid & 31;
  int half = lane >> 4, lm = lane & 15;
  long mt = (long)blockIdx.x * 8 + wave;
  bool active = (mt * 16) < (long)M;
  int m0 = (int)(mt << 4);
  const float* xrow = nullptr;
  if (active) {
    int r = m0 + lm;
    int bidx = r / Lout, l = r - bidx * Lout;
    xrow = X + ((long)bidx * Lin + l) * Cin;
  }
  v8f zero = {0.f,0.f,0.f,0.f,0.f,0.f,0.f,0.f};
  v8f acc[NT];
  #pragma unroll
  for (int nt = 0; nt < NT; ++nt) acc[nt] = zero;

  for (int t = 0; t < T; ++t) {
    int tot = Cin * Cout;
    for (int i = tid; i < tot; i += 256) {
      int co = i / Cin, ci = i - co * Cin;
      smem[ci * Cout + co] = Wt[((long)co * Cin + ci) * T + t];
    }
    __syncthreads();
    if (active) {
      const float2* xp = (const float2*)(xrow + (long)t * Cin);
      for (int k = 0; k < Cin; k += 4) {
        int ka = k + (half << 1);
        float2 av = xp[ka >> 1];
        v2f a; a.x = av.x; a.y = av.y;
        #pragma unroll
        for (int nt = 0; nt < NT; ++nt) {
          int n = (nt << 4) + lm;
          v2f b; b.x = smem[ka * Cout + n]; b.y = smem[(ka + 1) * Cout + n];
          acc[nt] = wmma_f32_k4(a, b, acc[nt]);
        }
      }
    }
    __syncthreads();
  }
  if (!active) return;
  #pragma unroll
  for (int v = 0; v < 8; ++v) {
    int row = m0 + v + (half << 3);
    #pragma unroll
    for (int nt = 0; nt < NT; ++nt) {
      int n = (nt << 4) + lm;
      Y[(long)row * Cout + n] = fmaxf(acc[nt][v] + bias[n], 0.0f);
    }
  }
}

// ---------------- GCN edge scatter ----------------
// out[dst,f] += hw[src,f] * dinv[src]*dinv[dst]   (32 threads/edge, float4 each)
__global__ void k_gcn_edges(const float* __restrict__ hw, const float* __restrict__ dinv,
                            const int* __restrict__ src, const int* __restrict__ dst,
                            float* __restrict__ out, int ldo) {
  long t = (long)blockIdx.x * blockDim.x + threadIdx.x;
  if (t >= (long)NE * 32) return;
  int e = (int)(t >> 5), q = (int)(t & 31);
  int s = src[e], d = dst[e];
  float nrm = dinv[s] * dinv[d];
  float4 v = ((const float4*)hw)[(long)s * 32 + q];
  float* o = out + (long)d * ldo + (q << 2);
  atomicAdd(o + 0, v.x * nrm);
  atomicAdd(o + 1, v.y * nrm);
  atomicAdd(o + 2, v.z * nrm);
  atomicAdd(o + 3, v.w * nrm);
}

__global__ void k_relu_strided(float* p, int ldo) {
  long i = (long)blockIdx.x * blockDim.x + threadIdx.x;
  if (i >= (long)NND * 128) return;
  int r = (int)(i >> 7), f = (int)(i & 127);
  float* q = p + (long)r * ldo + f;
  *q = fmaxf(*q, 0.0f);
}

// ---------------- discriminator ----------------
__global__ void k_colmax_part(const float* __restrict__ h, float* __restrict__ part, int chunk) {
  int t = threadIdx.x;                       // 256 cols
  int r0 = blockIdx.x * chunk;
  int r1 = r0 + chunk; if (r1 > NND) r1 = NND;
  float m = -3.4e38f;
  for (int r = r0; r < r1; ++r) m = fmaxf(m, h[(long)r * 256 + t]);
  part[(long)blockIdx.x * 256 + t] = m;
}

__global__ void k_colmax_final(const float* __restrict__ part, float* __restrict__ c, int nb) {
  int t = threadIdx.x;
  float m = -3.4e38f;
  for (int i = 0; i < nb; ++i) m = fmaxf(m, part[(long)i * 256 + t]);
  c[t] = 1.0f / (1.0f + __expf(-m));
}

__global__ void k_matvec256(const float* __restrict__ W, const float* __restrict__ c,
                            float* __restrict__ cw) {
  int i = threadIdx.x;
  float s = 0.f;
  for (int j = 0; j < 256; ++j) s += W[(long)i * 256 + j] * c[j];
  cw[i] = s;
}

__global__ void k_disc_scores(const float* __restrict__ hp, const float* __restrict__ hm,
                              const float* __restrict__ cw, const float* __restrict__ db,
                              float* __restrict__ out) {
  int wave = threadIdx.x >> 5, lane = threadIdx.x & 31;
  int node = blockIdx.x * 8 + wave;
  if (node >= NND) return;
  const float* hpr = hp + (long)node * 256;
  const float* hmr = hm + (long)node * 256;
  float s1 = 0.f, s2 = 0.f;
  #pragma unroll
  for (int s = 0; s < 8; ++s) {
    int j = lane + (s << 5);
    float w = cw[j];
    s1 += hpr[j] * w; s2 += hmr[j] * w;
  }
  #pragma unroll
  for (int off = 16; off; off >>= 1) {
    s1 += __shfl_xor(s1, off);
    s2 += __shfl_xor(s2, off);
  }
  if (lane == 0) {
    float b = db[0];
    out[(long)node * 2 + 0] = s1 + b;
    out[(long)node * 2 + 1] = s2 + b;
  }
}

// ---------------- sequence branches ----------------
__global__ void k_embed(const int* __restrict__ tok, const float* __restrict__ table,
                        float* __restrict__ out, long n, int C) {
  long i = (long)blockIdx.x * blockDim.x + threadIdx.x;
  if (i >= n * C) return;
  long p = i / C; int c = (int)(i - p * C);
  out[i] = table[(long)tok[p] * C + c];
}

__global__ void k_maxpool(const float* __restrict__ Y, float* __restrict__ pair,
                          int Lout, int coff) {
  int b = blockIdx.x, c = threadIdx.x;   // 128 threads
  float m = -3.4e38f;
  for (int l = 0; l < Lout; ++l) m = fmaxf(m, Y[((long)b * Lout + l) * 128 + c]);
  pair[(long)b * 256 + coff + c] = m;
}

__global__ void k_cls_final(const float* __restrict__ clsh, const float* __restrict__ w2,
                            float* __restrict__ out) {
  int t = threadIdx.x;                   // 256 threads
  int b = t >> 1, j = t & 1;
  float s = 0.f;
  for (int k = 0; k < 128; ++k) s += clsh[(long)b * 128 + k] * w2[k * 2 + j];
  out[((long)NND + b) * 2 + j] = s;
}

// ---------------- workspace layout (float offsets) ----------------
static const long OFF_DINV  = 0;                          // 50000 (padded 50176)
static const long OFF_BUF1  = 50176;                      // 12.8M  h_pl  | prot emb / y2
static const long OFF_BUF2  = OFF_BUF1 + 12800000;        // 12.8M  h_mi  | (emb spill)
static const long OFF_BUF3  = OFF_BUF2 + 12800000;        // 12.8M  cat   | prot y1
static const long OFF_BUF4  = OFF_BUF3 + 12800000;        // 6.4M   hw    | drug bufs
static const long OFF_SMALL = OFF_BUF4 + 6400000;
static const long OFF_PART  = OFF_SMALL;                  // 128*256
static const long OFF_C     = OFF_PART + 32768;           // 256
static const long OFF_CW    = OFF_C + 256;                // 256
static const long OFF_PAIR  = OFF_CW + 256;               // 128*256
static const long OFF_CLSH  = OFF_PAIR + 32768;           // 128*128

extern "C" void kernel_launch(void* const* d_in, const int* in_sizes, int n_in,
                              void* d_out, int out_size, void* d_ws, size_t ws_size,
                              hipStream_t stream) {
  const float* x        = (const float*)d_in[0];
  const int*   ei       = (const int*)d_in[1];
  const int*   perm     = (const int*)d_in[2];
  const int*   dtok     = (const int*)d_in[3];
  const int*   ptok     = (const int*)d_in[4];
  const float* W_o1 = (const float*)d_in[5];  const float* b_o1 = (const float*)d_in[6];
  const float* W_s1 = (const float*)d_in[7];  const float* b_s1 = (const float*)d_in[8];
  const float* W_o2 = (const float*)d_in[9];  const float* b_o2 = (const float*)d_in[10];
  const float* W_s2 = (const float*)d_in[11]; const float* b_s2 = (const float*)d_in[12];
  const float* discW = (const float*)d_in[13]; const float* discb = (const float*)d_in[14];
  const float* demb_t = (const float*)d_in[15];
  const float* dcw1 = (const float*)d_in[16]; const float* dcb1 = (const float*)d_in[17];
  const float* dcw2 = (const float*)d_in[18]; const float* dcb2 = (const float*)d_in[19];
  const float* pemb_t = (const float*)d_in[20];
  const float* pcw1 = (const float*)d_in[21]; const float* pcb1 = (const float*)d_in[22];
  const float* pcw2 = (const float*)d_in[23]; const float* pcb2 = (const float*)d_in[24];
  const float* clsw1 = (const float*)d_in[25]; const float* clsb1 = (const float*)d_in[26];
  const float* clsw2 = (const float*)d_in[27];

  float* ws   = (float*)d_ws;
  float* dinv = ws + OFF_DINV;
  float* hpl  = ws + OFF_BUF1;
  float* hmi  = ws + OFF_BUF2;
  float* cat  = ws + OFF_BUF3;
  float* hw   = ws + OFF_BUF4;
  float* out  = (float*)d_out;

  const int* src = ei;
  const int* dst = ei + NE;

  // ---- degrees -> dinv ----
  k_fill<<<(NND + 255) / 256, 256, 0, stream>>>(dinv, NND, 1.0f);
  k_edge_deg<<<(NE + 255) / 256, 256, 0, stream>>>(dst, dinv, NE);
  k_rsqrt<<<(NND + 255) / 256, 256, 0, stream>>>(dinv, NND);

  long nf = (long)NND * 128;
  int nfb = (int)((nf + 255) / 256);
  int efb = (int)(((long)NE * 32 + 255) / 256);
  int gemmBlocks = (NND / 16 + 7) / 8;   // 391 (last block: 5 active waves)
  const size_t LDS64K = 64 * 1024;

  // one gcn layer: {hw, o=hw*selfnorm+b} = gemm ; scatter ; optional relu
  auto gcn = [&](const float* A, int lda, const int* ridx, const float* W, int K,
                 const float* bias, float* o, int ldo, int relu) {
    k_wmma_gemm_strip<8><<<gemmBlocks, 256, LDS64K, stream>>>(
        A, lda, ridx, W, hw, o, ldo, dinv, bias, 0, K, NND);
    k_gcn_edges<<<efb, 256, 0, stream>>>(hw, dinv, src, dst, o, ldo);
    if (relu) k_relu_strided<<<nfb, 256, 0, stream>>>(o, ldo);
  };

  auto encode = [&](const int* ridx, float* h) {
    gcn(x, DD, ridx, W_o1, 128, b_o1, cat + 0,   256, 1);   // h_o -> cat[:, :128]
    gcn(x, DD, ridx, W_s1, 128, b_s1, cat + 128, 256, 1);   // h_s -> cat[:,128:]
    gcn(cat, 256, nullptr, W_o2, 256, b_o2, h + 0,   256, 0);
    gcn(cat, 256, nullptr, W_s2, 256, b_s2, h + 128, 256, 0);
  };

  encode(nullptr, hpl);   // positive
  encode(perm,    hmi);   // negative (shuffled features)

  // ---- discriminator ----
  int chunk = (NND + 127) / 128;
  k_colmax_part<<<128, 256, 0, stream>>>(hpl, ws + OFF_PART, chunk);
  k_colmax_final<<<1, 256, 0, stream>>>(ws + OFF_PART, ws + OFF_C, 128);
  k_matvec256<<<1, 256, 0, stream>>>(discW, ws + OFF_C, ws + OFF_CW);
  k_disc_scores<<<NND / 8, 256, 0, stream>>>(hpl, hmi, ws + OFF_CW, discb, out);

  // ---- drug branch (scratch aliases hw region; graph phase done) ----
  float* d_emb = hw;                 //  819200 floats
  float* d_y1  = hw + 1000000;       //  811008
  float* d_y2  = hw + 2000000;       // 1572864
  k_embed<<<(int)(((long)BB * 100 * 64 + 255) / 256), 256, 0, stream>>>(
      dtok, demb_t, d_emb, (long)BB * 100, 64);
  k_wmma_conv_strip<4><<<99, 256, 64 * 64 * 4, stream>>>(
      d_emb, dcw1, dcb1, d_y1, 100, 99, 64, 2, BB * 99);
  k_wmma_conv_strip<8><<<96, 256, 64 * 128 * 4, stream>>>(
      d_y1, dcw2, dcb2, d_y2, 99, 96, 64, 4, BB * 96);
  k_maxpool<<<BB, 128, 0, stream>>>(d_y2, ws + OFF_PAIR, 96, 0);

  // ---- protein branch (emb aliases h_pl/h_mi; y1 aliases cat; y2 re-aliases emb) ----
  float* p_emb = hpl;                // 16384000 (spans buf1+buf2)
  float* p_y1  = cat;                //  8183808
  float* p_y2  = hpl;                // emb dead once conv1 finishes; conv2 reads y1 only
  k_embed<<<(int)(((long)BB * 1000 * 128 + 255) / 256), 256, 0, stream>>>(
      ptok, pemb_t, p_emb, (long)BB * 1000, 128);
  k_wmma_conv_strip<4><<<999, 256, 128 * 64 * 4, stream>>>(
      p_emb, pcw1, pcb1, p_y1, 1000, 999, 128, 2, BB * 999);
  k_wmma_conv_strip<8><<<996, 256, 64 * 128 * 4, stream>>>(
      p_y1, pcw2, pcb2, p_y2, 999, 996, 64, 4, BB * 996);
  k_maxpool<<<BB, 128, 0, stream>>>(p_y2, ws + OFF_PAIR, 996, 128);

  // ---- classifier ----
  k_wmma_gemm_strip<8><<<1, 256, LDS64K, stream>>>(
      ws + OFF_PAIR, 256, nullptr, clsw1, nullptr, ws + OFF_CLSH, 128,
      nullptr, clsb1, 1, 256, BB);
  k_cls_final<<<1, 256, 0, stream>>>(ws + OFF_CLSH, clsw2, out);
}